// MemoryLayer_44650480009843
// MI455X (gfx1250) — compile-verified
//
#include <hip/hip_runtime.h>
#include <cstddef>

#define Bsz 32
#define Usz 512
#define Csz 4096
#define DKsz 32
#define DVsz 32
#define CT 128
#define NIT (Csz / CT)
#define MEM_RATE 0.01f
#define REWARD_RATE 0.01f
#define NEGV (-1e9f)

typedef __attribute__((ext_vector_type(2))) float v2f;
typedef __attribute__((ext_vector_type(8))) float v8f;

__device__ __forceinline__ v8f wmma4(v2f a, v2f b, v8f c) {
  // V_WMMA_F32_16X16X4_F32 : D = A(16x4) * B(4x16) + C(16x16), all f32
  return __builtin_amdgcn_wmma_f32_16x16x4_f32(false, a, false, b,
                                               (short)0, c, false, false);
}

__global__ __launch_bounds__(256) void memlayer_kernel(
    const float* __restrict__ att,  const float* __restrict__ val,
    const float* __restrict__ rew,  const int*   __restrict__ mask,
    const float* __restrict__ Watt, const float* __restrict__ temp,
    const float* __restrict__ kmem, const float* __restrict__ vmem,
    const float* __restrict__ rmem,
    float* __restrict__ out_o, float* __restrict__ out_w,
    float* __restrict__ out_k, float* __restrict__ out_v,
    float* __restrict__ out_r)
{
  const int u    = blockIdx.x;
  const int t    = threadIdx.x;
  const int lane = t & 31;          // wave32
  const int wave = t >> 5;          // 8 waves
  const int g    = lane >> 4;       // lane half-group
  const int r    = lane & 15;

  __shared__ float q_sm[32 * 33];     // query [b][k] (raw, un-scaled)
  __shared__ float kb_sm[CT * 33];    // key tile / value_mem tile / Watt staging
  __shared__ float lw_sm[32 * 132];   // logits tile / w tile [b][c]
  __shared__ float av_sm[32 * 33];    // attention staging, then value[:,u,:]
  __shared__ float rm_sm[CT];
  __shared__ float cs_sm[CT];
  __shared__ float csr_sm[CT];
  __shared__ float mrun[32], drun[32], mnew[32], dinv[32];
  __shared__ float redm[32 * 8], reds[32 * 8];
  __shared__ float rw_sm[32];
  __shared__ float red_sm[8 * 256];

  const float invT = 1.0f / temp[u];

  // ---- stage attention[:,u,:] and W_att[u] ----
  for (int i = t; i < 32 * 32; i += 256) {
    int b = i >> 5, d = i & 31;
    av_sm[b * 33 + d] = att[((size_t)b * Usz + u) * DKsz + d];
    kb_sm[b * 33 + d] = Watt[(size_t)u * DKsz * DKsz + i];   // [d][k]
  }
  if (t < 32) { mrun[t] = -3.0e38f; drun[t] = 0.0f; rw_sm[t] = rew[t]; }
  __syncthreads();

  // ---- G1: query = att @ W_att  (M=32,N=32,K=32), waves 0..3 ----
  if (wave < 4) {
    const int m0 = (wave >> 1) * 16, n0 = (wave & 1) * 16;
    v8f acc = {};
#pragma unroll
    for (int kk = 0; kk < 8; ++kk) {
      const int k0 = 4 * kk + 2 * g;
      v2f a, b;
      a.x = av_sm[(m0 + r) * 33 + k0];
      a.y = av_sm[(m0 + r) * 33 + k0 + 1];
      b.x = kb_sm[k0 * 33 + n0 + r];
      b.y = kb_sm[(k0 + 1) * 33 + n0 + r];
      acc = wmma4(a, b, acc);
    }
#pragma unroll
    for (int v = 0; v < 8; ++v)
      q_sm[(m0 + v + 8 * g) * 33 + n0 + r] = acc[v];
  }
  __syncthreads();

  // =======================  PHASE A  =======================
  for (int ci = 0; ci < NIT; ++ci) {
    const int c0 = ci * CT;
    for (int i = t; i < CT * 32; i += 256) {
      int c = i >> 5, k = i & 31;
      kb_sm[c * 33 + k] = kmem[((size_t)u * Csz + c0 + c) * DKsz + k];
    }
    if (t < CT) rm_sm[t] = rmem[(size_t)u * Csz + c0 + t];
    __syncthreads();

    // logits tile = query @ key^T ; each wave: n0=16*wave, both m-tiles
    {
      const int n0 = 16 * wave;
#pragma unroll
      for (int mt = 0; mt < 2; ++mt) {
        const int m0 = mt * 16;
        v8f acc = {};
#pragma unroll
        for (int kk = 0; kk < 8; ++kk) {
          const int k0 = 4 * kk + 2 * g;
          v2f a, b;
          a.x = q_sm[(m0 + r) * 33 + k0];
          a.y = q_sm[(m0 + r) * 33 + k0 + 1];
          b.x = kb_sm[(n0 + r) * 33 + k0];       // key^T from LDS
          b.y = kb_sm[(n0 + r) * 33 + k0 + 1];
          acc = wmma4(a, b, acc);
        }
#pragma unroll
        for (int v = 0; v < 8; ++v) {
          const int bb = m0 + v + 8 * g;
          const int cc = n0 + r;
          const float l = acc[v] * invT;
          const int keep = mask[((size_t)bb * Usz + u) * Csz + c0 + cc];
          lw_sm[bb * 132 + cc] = keep ? l : NEGV;
        }
      }
    }
    __syncthreads();

    // spill raw (masked, temp-scaled) logits into w buffer as scratch
    for (int i = t; i < 32 * CT; i += 256) {
      int b = i >> 7, c = i & 127;
      out_w[((size_t)b * Usz + u) * Csz + c0 + c] = lw_sm[b * 132 + c];
    }

    // online running max + reward-weighted denominator
    {
      const int b = t >> 3, j = t & 7;
      float mx = -3.0e38f;
#pragma unroll
      for (int e = 0; e < 16; ++e)
        mx = fmaxf(mx, lw_sm[b * 132 + j * 16 + e]);
      redm[b * 8 + j] = mx;
    }
    __syncthreads();
    if (t < 32) {
      float mx = mrun[t];
#pragma unroll
      for (int j = 0; j < 8; ++j) mx = fmaxf(mx, redm[t * 8 + j]);
      mnew[t] = mx;
    }
    __syncthreads();
    {
      const int b = t >> 3, j = t & 7;
      const float mn = mnew[b];
      float s = 0.0f;
#pragma unroll
      for (int e = 0; e < 16; ++e) {
        const int c = j * 16 + e;
        s += __expf(lw_sm[b * 132 + c] - mn) * rm_sm[c];
      }
      reds[b * 8 + j] = s;
    }
    __syncthreads();
    if (t < 32) {
      float d = drun[t] * __expf(mrun[t] - mnew[t]);
#pragma unroll
      for (int j = 0; j < 8; ++j) d += reds[t * 8 + j];
      drun[t] = d;
      mrun[t] = mnew[t];
    }
    __syncthreads();
  }
  if (t < 32) dinv[t] = 1.0f / drun[t];
  __threadfence();            // make logit scratch visible for Phase B reads
  __syncthreads();

  // =======================  PHASE B  =======================
  for (int i = t; i < 32 * 32; i += 256) {  // stage value[:,u,:]
    int b = i >> 5, v = i & 31;
    av_sm[b * 33 + v] = val[((size_t)b * Usz + u) * DVsz + v];
  }

  v8f oacc = {};                         // outputs partial
  const int ot = wave >> 1, oh = wave & 1;
  const int om0 = (ot >> 1) * 16, on0 = (ot & 1) * 16;

  for (int ci = 0; ci < NIT; ++ci) {
    const int c0 = ci * CT;
    if (t < CT) rm_sm[t] = rmem[(size_t)u * Csz + c0 + t];
    __syncthreads();

    // finalize w tile: w = exp(l-m)*rm/denom ; write out, keep in LDS
    for (int i = t; i < 32 * CT; i += 256) {
      int b = i >> 7, c = i & 127;
      const size_t ga = ((size_t)b * Usz + u) * Csz + c0 + c;
      const float l = out_w[ga];
      const float wv = __expf(l - mrun[b]) * rm_sm[c] * dinv[b];
      lw_sm[b * 132 + c] = wv;
      out_w[ga] = wv;
    }
    for (int i = t; i < CT * 32; i += 256) {   // stage value_mem tile
      int c = i >> 5, k = i & 31;
      kb_sm[c * 33 + k] = vmem[((size_t)u * Csz + c0 + c) * DVsz + k];
    }
    __syncthreads();

    // column sums over b + reward_mem update
    if (t < CT) {
      float s = 0.0f, sr = 0.0f;
#pragma unroll
      for (int b = 0; b < 32; ++b) {
        const float wv = lw_sm[b * 132 + t];
        s += wv;
        sr += wv * rw_sm[b];
      }
      cs_sm[t]  = s;
      csr_sm[t] = sr;
      out_r[(size_t)u * Csz + c0 + t] =
          rm_sm[t] * (1.0f - REWARD_RATE * s) + REWARD_RATE * sr;
    }
    __syncthreads();

    // outputs += w_tile @ value_mem_tile  (wave: tile ot, K-half oh)
#pragma unroll
    for (int kk = oh * 16; kk < oh * 16 + 16; ++kk) {
      const int k0 = 4 * kk + 2 * g;
      v2f a, b;
      a.x = lw_sm[(om0 + r) * 132 + k0];
      a.y = lw_sm[(om0 + r) * 132 + k0 + 1];
      b.x = kb_sm[k0 * 33 + on0 + r];
      b.y = kb_sm[(k0 + 1) * 33 + on0 + r];
      oacc = wmma4(a, b, oacc);
    }

    // key/value memory EMA updates: fresh^T @ {query,value} + old*retain
#pragma unroll
    for (int w2 = 0; w2 < 2; ++w2) {
      const int tt  = wave + w2 * 8;
      const int mc0 = (tt & 7) * 16;      // c offset inside tile
      const int n0  = (tt >> 3) * 16;     // k/v dim offset
      v8f dk = {}, dv = {};
#pragma unroll
      for (int kk = 0; kk < 8; ++kk) {
        const int k0 = 4 * kk + 2 * g;
        v2f a, bq, bv;
        a.x  = MEM_RATE * lw_sm[k0 * 132 + mc0 + r];        // fresh^T
        a.y  = MEM_RATE * lw_sm[(k0 + 1) * 132 + mc0 + r];
        bq.x = q_sm[k0 * 33 + n0 + r];
        bq.y = q_sm[(k0 + 1) * 33 + n0 + r];
        bv.x = av_sm[k0 * 33 + n0 + r];
        bv.y = av_sm[(k0 + 1) * 33 + n0 + r];
        dk = wmma4(a, bq, dk);
        dv = wmma4(a, bv, dv);
      }
#pragma unroll
      for (int v = 0; v < 8; ++v) {
        const int cc = mc0 + v + 8 * g;
        const float ret = 1.0f - MEM_RATE * cs_sm[cc];
        const size_t ga = ((size_t)u * Csz + c0 + cc) * DKsz + n0 + r;
        out_k[ga] = kmem[ga] * ret + dk[v];
        out_v[ga] = vmem[ga] * ret + dv[v];
      }
    }
    __syncthreads();
  }

  // reduce outputs partials across K-half wave pairs
#pragma unroll
  for (int v = 0; v < 8; ++v)
    red_sm[wave * 256 + lane * 8 + v] = oacc[v];
  __syncthreads();
  if (oh == 0) {
#pragma unroll
    for (int v = 0; v < 8; ++v) {
      const float s = red_sm[wave * 256 + lane * 8 + v] +
                      red_sm[(wave + 1) * 256 + lane * 8 + v];
      const int bb = om0 + v + 8 * g;
      out_o[((size_t)bb * Usz + u) * DVsz + on0 + r] = s;
    }
  }
}

extern "C" void kernel_launch(void* const* d_in, const int* in_sizes, int n_in,
                              void* d_out, int out_size, void* d_ws, size_t ws_size,
                              hipStream_t stream) {
  const float* att  = (const float*)d_in[0];
  const float* val  = (const float*)d_in[1];
  const float* rew  = (const float*)d_in[2];
  const int*   mask = (const int*)d_in[3];
  const float* Watt = (const float*)d_in[4];
  const float* temp = (const float*)d_in[5];
  const float* kmem = (const float*)d_in[6];
  const float* vmem = (const float*)d_in[7];
  const float* rmem = (const float*)d_in[8];

  float* out   = (float*)d_out;
  float* out_o = out;                                      // [B,U,DV]
  float* out_w = out_o + (size_t)Bsz * Usz * DVsz;         // [B,U,C]
  float* out_k = out_w + (size_t)Bsz * Usz * Csz;          // [U,C,DK]
  float* out_v = out_k + (size_t)Usz * Csz * DKsz;         // [U,C,DV]
  float* out_r = out_v + (size_t)Usz * Csz * DVsz;         // [U,C]

  memlayer_kernel<<<Usz, 256, 0, stream>>>(att, val, rew, mask, Watt, temp,
                                           kmem, vmem, rmem,
                                           out_o, out_w, out_k, out_v, out_r);
}